// KNNFunction_6828998001133
// MI455X (gfx1250) — compile-verified
//
#include <hip/hip_runtime.h>
#include <hip/hip_bf16.h>

// CDNA5 / gfx1250 fused KNN (squared-L2, top-16) using V_WMMA_F32_16X16X4_F32.
//
// Distance trick: A[m,0..2] = p2[m].xyz, A[m,3] = |p2[m]|^2
//                 B[0..2,j] = -2*p1[j].xyz, B[3,j] = 1
//   => D[m,j] = |p2_m|^2 - 2<p1_j, p2_m>    (add |p1_j|^2 at the very end;
//      it is constant per lane so it does not affect the top-K ordering).

typedef __attribute__((ext_vector_type(2))) float v2f;
typedef __attribute__((ext_vector_type(8))) float v8f;

#define KNN_K          16
#define KNN_N          4
#define KNN_D          3
#define WAVES_PB       8
#define QPW            16                 // queries per wave
#define QPB            (WAVES_PB * QPW)   // 128 queries per block

// Branchless sorted insert: one v_cmp + 4 v_cndmask per stage, no EXEC branches.
__device__ __forceinline__ void topk_insert(float (&bd)[KNN_K], int (&bi)[KNN_K],
                                            float v, int vi)
{
#pragma unroll
    for (int s = 0; s < KNN_K; ++s) {
        const bool lt = v < bd[s];
        const float nb = lt ? v     : bd[s];
        const float nv = lt ? bd[s] : v;
        const int   ni = lt ? vi    : bi[s];
        const int   nc = lt ? bi[s] : vi;
        bd[s] = nb;  v  = nv;
        bi[s] = ni;  vi = nc;
    }
}

// One tile's worth of selection work on the 8 accumulator rows.
__device__ __forceinline__ void process_tile(const v8f& acc, int cbase,
                                             float (&bd)[KNN_K], int (&bi)[KNN_K])
{
#pragma unroll
    for (int r = 0; r < 8; ++r) {
        const float d = acc[r];
        if (d < bd[KNN_K - 1]) {          // single guard; insert is branchless
            topk_insert(bd, bi, d, cbase + r);
        }
    }
}

__global__ __launch_bounds__(256)
void knn_wmma_kernel(const float* __restrict__ p1,
                     const float* __restrict__ p2,
                     int*   __restrict__ out_idx,
                     float* __restrict__ out_dist,
                     int P1, int P2)
{
    extern __shared__ char smem_raw[];
    float4* lds_p2 = (float4*)smem_raw;           // [P2] : (x, y, z, |p|^2)

    const int tid  = threadIdx.x;
    const int lane = tid & 31;
    const int wave = tid >> 5;
    const int hf   = lane >> 4;     // lane half: selects K-pair (0,1) vs (2,3)
    const int j    = lane & 15;     // query column within wave tile / cand row of A

    const int blocksPerBatch = P1 / QPB;
    const int n     = blockIdx.x / blocksPerBatch;
    const int chunk = blockIdx.x % blocksPerBatch;

    // ---- Stage p2[n] into LDS as (x,y,z,sq); 4 points via 3x b128 loads ----
    const float* p2n = p2 + (size_t)n * P2 * KNN_D;
    for (int p4 = tid; p4 < (P2 >> 2); p4 += blockDim.x) {
        const float4* src = (const float4*)(p2n + (size_t)p4 * 12);
        float4 a = src[0], b = src[1], c = src[2];
        const int p = p4 * 4;
        lds_p2[p + 0] = make_float4(a.x, a.y, a.z, a.x*a.x + a.y*a.y + a.z*a.z);
        lds_p2[p + 1] = make_float4(a.w, b.x, b.y, a.w*a.w + b.x*b.x + b.y*b.y);
        lds_p2[p + 2] = make_float4(b.z, b.w, c.x, b.z*b.z + b.w*b.w + c.x*c.x);
        lds_p2[p + 3] = make_float4(c.y, c.z, c.w, c.y*c.y + c.z*c.z + c.w*c.w);
    }
    __syncthreads();

    // ---- Loop-invariant B matrix (4x16): per-lane K-pair ------------------
    // B VGPR0 -> K = 2*hf, VGPR1 -> K = 2*hf+1  (lane-half covers half of K)
    const int qi = chunk * QPB + wave * QPW + j;
    const float* p1q = p1 + ((size_t)n * P1 + qi) * KNN_D;
    const float qx = p1q[0], qy = p1q[1], qz = p1q[2];
    const float sq1 = qx * qx + qy * qy + qz * qz;
    v2f B;
    B.x = hf ? (-2.0f * qz) : (-2.0f * qx);
    B.y = hf ? 1.0f         : (-2.0f * qy);

    // ---- Per-lane sorted top-K (this lane sees candidates with
    //      (c mod 16) in [8*hf, 8*hf+8) ) ----------------------------------
    float bd[KNN_K];
    int   bi[KNN_K];
#pragma unroll
    for (int s = 0; s < KNN_K; ++s) { bd[s] = 3.0e38f; bi[s] = 0; }

    // A-tile fetch, software-pipelined one iteration ahead.
    // Element (tile*16 + j)*2 + hf -> byte offset tile*256 + (j*2+hf)*8.
    const char* aptr = smem_raw + (size_t)(j * 2 + hf) * 8;
    const int ntiles = P2 >> 4;

    float2 af = *(const float2*)aptr;             // tile 0
    aptr += 256;
    int cbase = 8 * hf;                           // D VGPR r -> candidate cbase + r

#pragma unroll 1
    for (int jt = 0; jt < ntiles - 1; ++jt) {
        v2f A; A.x = af.x; A.y = af.y;
        af = *(const float2*)aptr;                // prefetch tile jt+1
        aptr += 256;

        v8f acc = {};
        acc = __builtin_amdgcn_wmma_f32_16x16x4_f32(
                  /*neg_a=*/false, A, /*neg_b=*/false, B,
                  /*c_mod=*/(short)0, acc, /*reuse_a=*/false, /*reuse_b=*/false);

        process_tile(acc, cbase, bd, bi);
        cbase += 16;
    }
    {   // last tile (no prefetch)
        v2f A; A.x = af.x; A.y = af.y;
        v8f acc = {};
        acc = __builtin_amdgcn_wmma_f32_16x16x4_f32(
                  false, A, false, B, (short)0, acc, false, false);
        process_tile(acc, cbase, bd, bi);
    }

    // ---- Merge the two half-lane lists (lanes j and j+16) -----------------
    __syncthreads();                              // done with p2 region; reuse it
    float* mD = (float*)smem_raw;                 // [8][32][17] padded rows
    int*   mI = (int*)(smem_raw + WAVES_PB * 32 * 17 * 4);
    {
        const int rowbase = (wave * 32 + lane) * 17;
#pragma unroll
        for (int s = 0; s < KNN_K; ++s) { mD[rowbase + s] = bd[s]; mI[rowbase + s] = bi[s]; }
    }
    __syncthreads();

    if (lane < 16) {
        const int ra = (wave * 32 + lane) * 17;
        const int rb = (wave * 32 + lane + 16) * 17;
        const size_t obase = ((size_t)n * P1 + qi) * KNN_K;
        int ia = 0, ib = 0;
#pragma unroll 1
        for (int k = 0; k < KNN_K; ++k) {
            float da = mD[ra + ia], db = mD[rb + ib];
            bool  ta = (da <= db);
            float dv = ta ? da : db;
            int   iv = ta ? mI[ra + ia] : mI[rb + ib];
            ia += ta ? 1 : 0;
            ib += ta ? 0 : 1;
            out_idx [obase + k] = iv;
            out_dist[obase + k] = dv + sq1;       // fold |p1_j|^2 back in
        }
    }
}

extern "C" void kernel_launch(void* const* d_in, const int* in_sizes, int n_in,
                              void* d_out, int out_size, void* d_ws, size_t ws_size,
                              hipStream_t stream)
{
    const float* p1 = (const float*)d_in[0];
    const float* p2 = (const float*)d_in[1];
    // d_in[2] is K (=16), hardcoded as KNN_K.

    const int P1 = in_sizes[0] / (KNN_N * KNN_D);
    const int P2 = in_sizes[1] / (KNN_N * KNN_D);

    // d_out: idxs (N*P1*K int32) then dists (N*P1*K float32), flat-concatenated.
    int*   out_idx  = (int*)d_out;
    float* out_dist = (float*)d_out + (size_t)KNN_N * P1 * KNN_K;

    const int blocks = KNN_N * (P1 / QPB);
    const size_t shmem = (size_t)P2 * sizeof(float4);   // 128 KB (merge reuses it)

    knn_wmma_kernel<<<blocks, 256, shmem, stream>>>(p1, p2, out_idx, out_dist, P1, P2);
}